// SimpleGCN_14328010899646
// MI455X (gfx1250) — compile-verified
//
#include <hip/hip_runtime.h>
#include <hip/hip_bf16.h>
#include <math.h>

typedef float v2f __attribute__((ext_vector_type(2)));
typedef float v8f __attribute__((ext_vector_type(8)));

// ---------------------------------------------------------------------------
// Degree / normalization (computed once; reused by all 3 layers)
// ---------------------------------------------------------------------------
__global__ __launch_bounds__(256) void deg_init(float* __restrict__ deg, int n) {
    int i = blockIdx.x * blockDim.x + threadIdx.x;
    if (i < n) deg[i] = 1.0f;                       // self-loop weight
}

__global__ __launch_bounds__(256) void deg_edges(const int* __restrict__ dst,
                                                 const float* __restrict__ ew,
                                                 float* __restrict__ deg, int E) {
    int e = blockIdx.x * blockDim.x + threadIdx.x;
    if (e < E) unsafeAtomicAdd(&deg[dst[e]], ew[e]);  // global_atomic_add_f32
}

__global__ __launch_bounds__(256) void deg_rsqrt(float* __restrict__ deg, int n) {
    int i = blockIdx.x * blockDim.x + threadIdx.x;
    if (i < n) {
        float v = deg[i];
        deg[i] = (v > 0.0f) ? rsqrtf(v) : 0.0f;     // becomes dinv[]
    }
}

// ---------------------------------------------------------------------------
// GEMM: H = X @ W using V_WMMA_F32_16X16X4_F32 (full f32 precision).
// One wave (32 threads) computes a 16-row stripe across all FOUT columns.
// f32 WMMA lane layouts (ISA 7.12.2):
//   A (16x4): lane l(<16)/l+16 holds row M=l; v0=K{0|2}, v1=K{1|3}
//   B (4x16): v0 = row K{0|2} across lanes, v1 = row K{1|3}
//   C/D     : vgpr r -> M = r + 8*half, N = lane%16
// ---------------------------------------------------------------------------
template<int FOUT>
__global__ __launch_bounds__(32) void gcn_gemm_wmma(const float* __restrict__ X,
                                                    const float* __restrict__ W,
                                                    float* __restrict__ H,
                                                    int n) {
    constexpr int K  = 128;
    constexpr int NT = FOUT / 16;

    const int row0 = blockIdx.x * 16;
    const int lane = threadIdx.x;       // 0..31 (wave32)
    const int hf   = lane >> 4;         // 0|1
    const int l    = lane & 15;

    int r = row0 + l;
    if (r > n - 1) r = n - 1;           // clamp (keeps EXEC all-ones for WMMA)
    const float* __restrict__ xrow = X + (size_t)r * K;

    v8f acc[NT];
#pragma unroll
    for (int t = 0; t < NT; ++t) acc[t] = v8f{0.f,0.f,0.f,0.f,0.f,0.f,0.f,0.f};

#pragma unroll 4
    for (int k0 = 0; k0 < K; k0 += 4) {
        const int ka = k0 + 2 * hf;
        v2f a;
        a.x = xrow[ka];
        a.y = xrow[ka + 1];
#pragma unroll
        for (int t = 0; t < NT; ++t) {
            const int col = t * 16 + l;
            v2f b;
            b.x = W[(size_t)ka * FOUT + col];
            b.y = W[(size_t)(ka + 1) * FOUT + col];
            acc[t] = __builtin_amdgcn_wmma_f32_16x16x4_f32(
                false, a, false, b, (short)0, acc[t], false, false);
        }
    }

#pragma unroll
    for (int t = 0; t < NT; ++t) {
#pragma unroll
        for (int rr = 0; rr < 8; ++rr) {
            const int m = row0 + rr + 8 * hf;
            if (m < n) H[(size_t)m * FOUT + t * 16 + l] = acc[t][rr];
        }
    }
}

// ---------------------------------------------------------------------------
// Self-loop init: out[i,:] = H[i,:] * dinv[i]^2   (norm of the self-loop edge)
// ---------------------------------------------------------------------------
template<int FOUT>
__global__ __launch_bounds__(256) void gcn_selfloop(const float* __restrict__ H,
                                                    const float* __restrict__ dinv,
                                                    float* __restrict__ out, int n) {
    int i = blockIdx.x * blockDim.x + threadIdx.x;
    if (i < n * FOUT) {
        float di = dinv[i / FOUT];
        out[i] = H[i] * di * di;
    }
}

// ---------------------------------------------------------------------------
// Edge scatter: out[dst,:] += H[src,:] * (dinv[src]*w*dinv[dst])
// FOUT/4 threads per edge, float4 gather + 4 hw float atomics.
// ---------------------------------------------------------------------------
template<int FOUT>
__global__ __launch_bounds__(256) void gcn_scatter(const float* __restrict__ H,
                                                   const int* __restrict__ src,
                                                   const int* __restrict__ dst,
                                                   const float* __restrict__ ew,
                                                   const float* __restrict__ dinv,
                                                   float* __restrict__ out, int E) {
    constexpr int TPE = FOUT / 4;       // threads per edge
    constexpr int EPB = 256 / TPE;      // edges per block
    const int lid = threadIdx.x;
    const int e = blockIdx.x * EPB + lid / TPE;
    if (e >= E) return;
    const int f4 = (lid % TPE) * 4;

    const int s = src[e];
    const int d = dst[e];
    const float norm = dinv[s] * ew[e] * dinv[d];

    const float4 h4 = *(const float4*)(H + (size_t)s * FOUT + f4);
    float* o = out + (size_t)d * FOUT + f4;
    unsafeAtomicAdd(o + 0, h4.x * norm);
    unsafeAtomicAdd(o + 1, h4.y * norm);
    unsafeAtomicAdd(o + 2, h4.z * norm);
    unsafeAtomicAdd(o + 3, h4.w * norm);
}

// ---------------------------------------------------------------------------
// Epilogue: out += bias; ACT 0 = ReLU, 1 = sigmoid
// ---------------------------------------------------------------------------
template<int FOUT, int ACT>
__global__ __launch_bounds__(256) void gcn_bias_act(float* __restrict__ out,
                                                    const float* __restrict__ b, int n) {
    int i = blockIdx.x * blockDim.x + threadIdx.x;
    if (i >= n * FOUT) return;
    float v = out[i] + b[i % FOUT];
    if (ACT == 0) v = fmaxf(v, 0.0f);
    else          v = 1.0f / (1.0f + expf(-v));
    out[i] = v;
}

// ---------------------------------------------------------------------------
// Host-side layer driver
// ---------------------------------------------------------------------------
template<int FOUT, int ACT>
static void run_layer(const float* X, const float* W, const float* b,
                      float* H, float* out, const float* dinv,
                      const int* src, const int* dst, const float* ew,
                      int n, int E, hipStream_t stream) {
    gcn_gemm_wmma<FOUT><<<(n + 15) / 16, 32, 0, stream>>>(X, W, H, n);
    const int total = n * FOUT;
    gcn_selfloop<FOUT><<<(total + 255) / 256, 256, 0, stream>>>(H, dinv, out, n);
    constexpr int EPB = 256 / (FOUT / 4);
    gcn_scatter<FOUT><<<(E + EPB - 1) / EPB, 256, 0, stream>>>(H, src, dst, ew, dinv, out, E);
    gcn_bias_act<FOUT, ACT><<<(total + 255) / 256, 256, 0, stream>>>(out, b, n);
}

extern "C" void kernel_launch(void* const* d_in, const int* in_sizes, int n_in,
                              void* d_out, int out_size, void* d_ws, size_t ws_size,
                              hipStream_t stream) {
    const float* x  = (const float*)d_in[0];   // [N,128]
    const int*   ei = (const int*)  d_in[1];   // [2,E]
    const float* ew = (const float*)d_in[2];   // [E]
    const float* W1 = (const float*)d_in[3];
    const float* b1 = (const float*)d_in[4];
    const float* W2 = (const float*)d_in[5];
    const float* b2 = (const float*)d_in[6];
    const float* W3 = (const float*)d_in[7];
    const float* b3 = (const float*)d_in[8];
    float* out = (float*)d_out;

    const int n = in_sizes[0] / 128;
    const int E = in_sizes[2];
    const int* src = ei;
    const int* dst = ei + E;

    // Workspace: dinv[n] | bufA[n*128] | bufB[n*128]  (~103 MB)
    char* ws = (char*)d_ws;
    const size_t off0 = ((size_t)n * sizeof(float) + 255) & ~(size_t)255;
    float* dinv = (float*)ws;
    float* bufA = (float*)(ws + off0);
    float* bufB = bufA + (size_t)n * 128;

    // Normalization (once for all layers)
    deg_init <<<(n + 255) / 256, 256, 0, stream>>>(dinv, n);
    deg_edges<<<(E + 255) / 256, 256, 0, stream>>>(dst, ew, dinv, E);
    deg_rsqrt<<<(n + 255) / 256, 256, 0, stream>>>(dinv, n);

    // Layer 1: x -> H(bufA) -> agg(bufB), ReLU
    run_layer<128, 0>(x,    W1, b1, bufA, bufB, dinv, src, dst, ew, n, E, stream);
    // Layer 2: bufB -> H(bufA) -> agg(bufB) (input dead after GEMM), ReLU
    run_layer<128, 0>(bufB, W2, b2, bufA, bufB, dinv, src, dst, ew, n, E, stream);
    // Layer 3: bufB -> H(bufA as N x 64) -> agg(d_out), sigmoid
    run_layer<64, 1>(bufB,  W3, b3, bufA, out,  dinv, src, dst, ew, n, E, stream);
}